// GAT_6451040878848
// MI455X (gfx1250) — compile-verified
//
#include <hip/hip_runtime.h>
#include <math.h>

#define NEG_SLOPE 0.2f

typedef float v2f __attribute__((ext_vector_type(2)));
typedef float v8f __attribute__((ext_vector_type(8)));

// ---------------------------------------------------------------------------
// WMMA f32 GEMM: C[M,64] = A[M,K] * B[K,64], row-major, K compile-time (64/128)
// Block = 128 threads (4 waves). Each wave owns a 16x16 output tile:
//   wave w -> columns [16w, 16w+16); block b -> rows [16b, 16b+16).
// V_WMMA_F32_16X16X4_F32 operand layout (ISA 7.12.2):
//   A 16x4:  lanes 0-15 = rows, VGPR0/1 = K0/K1; lanes 16-31 = K2/K3
//   B 4x16:  VGPR0: lanes0-15 K=0, lanes16-31 K=2; VGPR1: K=1 / K=3
//   C 16x16: VGPR j: lanes0-15 -> M=j, lanes16-31 -> M=j+8; N = lane&15
// ---------------------------------------------------------------------------
template <int K>
__global__ void gemm_wmma(const float* __restrict__ A,
                          const float* __restrict__ B,
                          float* __restrict__ C, int M) {
    const int lane  = threadIdx.x & 31;
    const int wave  = threadIdx.x >> 5;          // 0..3 -> column tile
    const int row0  = blockIdx.x * 16;
    const int col0  = wave * 16;
    const int r     = lane & 15;
    const int khalf = (lane >> 4) * 2;           // 0 or 2

    int arow = row0 + r; if (arow >= M) arow = M - 1;   // clamp (M % 16 == 0 normally)

    v8f acc = {};
#pragma unroll
    for (int k = 0; k < K; k += 4) {
        v2f a, b;
        const float* ap = A + (size_t)arow * K + (k + khalf);
        a.x = ap[0];
        a.y = ap[1];
        const float* bp = B + (size_t)(k + khalf) * 64 + col0 + r;
        b.x = bp[0];      // K = k+khalf
        b.y = bp[64];     // K = k+khalf+1
        acc = __builtin_amdgcn_wmma_f32_16x16x4_f32(
            false, a, false, b, (short)0, acc, false, false);
    }

    const int rbase = row0 + ((lane >> 4) << 3); // +8 for upper half-wave
    const int cc    = col0 + r;
#pragma unroll
    for (int j = 0; j < 8; ++j) {
        int rr = rbase + j;
        if (rr < M) C[(size_t)rr * 64 + cc] = acc[j];
    }
}

// ---------------------------------------------------------------------------
// Per-(node, head): alpha_src/alpha_dst dot products + init max/denom buffers
// ---------------------------------------------------------------------------
__global__ void node_alpha(const float* __restrict__ h,
                           const float* __restrict__ a_src,
                           const float* __restrict__ a_dst,
                           float* __restrict__ as, float* __restrict__ ad,
                           float* __restrict__ mx, float* __restrict__ dn,
                           int n, int heads, int ch) {
    int idx = blockIdx.x * blockDim.x + threadIdx.x;
    if (idx >= n * heads) return;
    int node = idx / heads;
    int hd   = idx % heads;
    const float* hp = h + (size_t)node * heads * ch + hd * ch;
    const float* ws = a_src + hd * ch;
    const float* wd = a_dst + hd * ch;
    float s = 0.f, d = 0.f;
    for (int c = 0; c < ch; ++c) {
        float v = hp[c];
        s += v * ws[c];
        d += v * wd[c];
    }
    as[idx] = s;
    ad[idx] = d;
    mx[idx] = -INFINITY;
    dn[idx] = 0.f;
}

__global__ void zero_f32(float* __restrict__ p, long n) {
    long i = (long)blockIdx.x * blockDim.x + threadIdx.x;
    if (i < n) p[i] = 0.f;
}

// sign-aware float atomic max (monotone int/uint mapping of IEEE floats)
__device__ inline void atomicMaxF(float* addr, float v) {
    if (v >= 0.f) atomicMax((int*)addr, __float_as_int(v));
    else          atomicMin((unsigned int*)addr, __float_as_uint(v));
}

__device__ inline void edge_endpoints(const int* __restrict__ ei, int e, int E,
                                      int& s, int& d) {
    if (e < E) { s = ei[e]; d = ei[E + e]; }
    else       { s = e - E; d = e - E; }       // self-loop
}

// pass A: segment max of leaky-relu(e) grouped by dst
__global__ void edge_max(const int* __restrict__ ei, int E, int ET,
                         const float* __restrict__ as,
                         const float* __restrict__ ad,
                         float* __restrict__ mx, int heads) {
    int idx = blockIdx.x * blockDim.x + threadIdx.x;
    if (idx >= ET * heads) return;
    int e = idx / heads, hd = idx % heads;
    int s, d; edge_endpoints(ei, e, E, s, d);
    float ev = as[s * heads + hd] + ad[d * heads + hd];
    ev = ev > 0.f ? ev : NEG_SLOPE * ev;
    atomicMaxF(&mx[d * heads + hd], ev);
}

// pass B: p = exp(e - max[dst]); store p; denom[dst] += p
__global__ void edge_exp(const int* __restrict__ ei, int E, int ET,
                         const float* __restrict__ as,
                         const float* __restrict__ ad,
                         const float* __restrict__ mx,
                         float* __restrict__ dn,
                         float* __restrict__ pbuf, int heads) {
    int idx = blockIdx.x * blockDim.x + threadIdx.x;
    if (idx >= ET * heads) return;
    int e = idx / heads, hd = idx % heads;
    int s, d; edge_endpoints(ei, e, E, s, d);
    float ev = as[s * heads + hd] + ad[d * heads + hd];
    ev = ev > 0.f ? ev : NEG_SLOPE * ev;
    float p = expf(ev - mx[d * heads + hd]);
    pbuf[idx] = p;
    atomicAdd(&dn[d * heads + hd], p);
}

// pass C: agg[dst, c] += h[src, c] * p/(denom[dst]+eps); thread per (edge, chan)
__global__ void edge_agg(const int* __restrict__ ei, int E, int ET,
                         const float* __restrict__ pbuf,
                         const float* __restrict__ dn,
                         const float* __restrict__ h,
                         float* __restrict__ agg, int heads, int ch) {
    int idx = blockIdx.x * blockDim.x + threadIdx.x;
    if (idx >= ET * 64) return;
    int e = idx >> 6;
    int c = idx & 63;
    int hd = c / ch;
    int s, d; edge_endpoints(ei, e, E, s, d);
    float p = pbuf[e * heads + hd];
    float alpha = p / (dn[d * heads + hd] + 1e-16f);
    atomicAdd(&agg[(size_t)d * 64 + c], h[(size_t)s * 64 + c] * alpha);
}

__global__ void bias_elu(const float* __restrict__ agg,
                         const float* __restrict__ b,
                         float* __restrict__ out, int n) {
    int idx = blockIdx.x * blockDim.x + threadIdx.x;
    if (idx >= n * 64) return;
    float v = agg[idx] + b[idx & 63];
    out[idx] = v > 0.f ? v : expf(v) - 1.f;   // ELU(alpha=1)
}

// +bias then log_softmax over 64 channels; one wave per node, shfl reductions
__global__ void finish_logsoftmax(const float* __restrict__ agg,
                                  const float* __restrict__ b,
                                  float* __restrict__ out, int n) {
    int wavesPerBlock = blockDim.x >> 5;
    int w = blockIdx.x * wavesPerBlock + (threadIdx.x >> 5);
    int lane = threadIdx.x & 31;
    if (w >= n) return;                        // whole wave exits together
    float v0 = agg[(size_t)w * 64 + lane]      + b[lane];
    float v1 = agg[(size_t)w * 64 + lane + 32] + b[lane + 32];
    float m = fmaxf(v0, v1);
#pragma unroll
    for (int o = 16; o > 0; o >>= 1) m = fmaxf(m, __shfl_xor(m, o, 32));
    float s = expf(v0 - m) + expf(v1 - m);
#pragma unroll
    for (int o = 16; o > 0; o >>= 1) s += __shfl_xor(s, o, 32);
    float lse = m + logf(s);
    out[(size_t)w * 64 + lane]      = v0 - lse;
    out[(size_t)w * 64 + lane + 32] = v1 - lse;
}

static inline int nblk(long n, int b) { return (int)((n + b - 1) / b); }

extern "C" void kernel_launch(void* const* d_in, const int* in_sizes, int n_in,
                              void* d_out, int out_size, void* d_ws, size_t ws_size,
                              hipStream_t stream) {
    const float* x    = (const float*)d_in[0];
    const int*   ei   = (const int*)d_in[1];
    const float* W1   = (const float*)d_in[2];
    const float* aS1  = (const float*)d_in[3];
    const float* aD1  = (const float*)d_in[4];
    const float* b1   = (const float*)d_in[5];
    const float* W2   = (const float*)d_in[6];
    const float* aS2  = (const float*)d_in[7];
    const float* aD2  = (const float*)d_in[8];
    const float* b2   = (const float*)d_in[9];
    float* out = (float*)d_out;

    const int N  = in_sizes[0] / 128;
    const int E  = in_sizes[1] / 2;
    const int ET = E + N;

    // workspace carve-up (floats)
    float* p    = (float*)d_ws;
    float* h    = p; p += (size_t)N * 64;   // layer features (reused for layer 2)
    float* agg  = p; p += (size_t)N * 64;   // aggregation target (reused)
    float* act  = p; p += (size_t)N * 64;   // ELU(layer-1 output)
    float* as1  = p; p += (size_t)N * 8;
    float* ad1  = p; p += (size_t)N * 8;
    float* mx1  = p; p += (size_t)N * 8;
    float* dn1  = p; p += (size_t)N * 8;
    float* as2  = p; p += (size_t)N;
    float* ad2  = p; p += (size_t)N;
    float* mx2  = p; p += (size_t)N;
    float* dn2  = p; p += (size_t)N;
    float* pbuf = p; p += (size_t)ET * 8;

    const int B = 256;

    // ---- layer 1 (8 heads x 8 ch, concat) ----
    gemm_wmma<128><<<nblk(N, 16), 128, 0, stream>>>(x, W1, h, N);
    node_alpha<<<nblk((long)N * 8, B), B, 0, stream>>>(h, aS1, aD1, as1, ad1, mx1, dn1, N, 8, 8);
    zero_f32<<<nblk((long)N * 64, B), B, 0, stream>>>(agg, (long)N * 64);
    edge_max<<<nblk((long)ET * 8, B), B, 0, stream>>>(ei, E, ET, as1, ad1, mx1, 8);
    edge_exp<<<nblk((long)ET * 8, B), B, 0, stream>>>(ei, E, ET, as1, ad1, mx1, dn1, pbuf, 8);
    edge_agg<<<nblk((long)ET * 64, B), B, 0, stream>>>(ei, E, ET, pbuf, dn1, h, agg, 8, 8);
    bias_elu<<<nblk((long)N * 64, B), B, 0, stream>>>(agg, b1, act, N);

    // ---- layer 2 (1 head x 64 ch, mean==identity) ----
    gemm_wmma<64><<<nblk(N, 16), 128, 0, stream>>>(act, W2, h, N);
    node_alpha<<<nblk((long)N, B), B, 0, stream>>>(h, aS2, aD2, as2, ad2, mx2, dn2, N, 1, 64);
    zero_f32<<<nblk((long)N * 64, B), B, 0, stream>>>(agg, (long)N * 64);
    edge_max<<<nblk((long)ET, B), B, 0, stream>>>(ei, E, ET, as2, ad2, mx2, 1);
    edge_exp<<<nblk((long)ET, B), B, 0, stream>>>(ei, E, ET, as2, ad2, mx2, dn2, pbuf, 1);
    edge_agg<<<nblk((long)ET * 64, B), B, 0, stream>>>(ei, E, ET, pbuf, dn2, h, agg, 1, 64);
    finish_logsoftmax<<<nblk((long)N, 8), 256, 0, stream>>>(agg, b2, out, N);
}